// RuchbahMoELayer_4131758538905
// MI455X (gfx1250) — compile-verified
//
#include <hip/hip_runtime.h>
#include <hip/hip_bf16.h>
#include <math.h>

typedef __attribute__((ext_vector_type(2))) float v2f;
typedef __attribute__((ext_vector_type(8))) float v8f;

#define H        1024
#define EXPERTS  64
#define WAVES    8
#define TOK_WG   128          // 8 waves * 16 tokens
#define KPAIRS   (H / 2)      // 512
// LDS float layout:
//   [0 .. 65535]            gate_w staged as [pair=k/2][e][parity]  (512*128)
//   [65536 .. 73727]        per-wave score scratch: 8 waves * 16 tok * 64 exp
//   [73728 .. 73855]        z^2 per token (128)
//   [73856 .. 73983]        inv(sum) per token (128)
#define LDS_B     0
#define LDS_S     65536
#define LDS_Z     (65536 + WAVES*16*EXPERTS)
#define LDS_INV   (LDS_Z + TOK_WG)
#define LDS_FLOATS (LDS_INV + TOK_WG)

__global__ __launch_bounds__(256) void moe_gate_kernel(
    const float* __restrict__ x,       // [T, 1024]
    const float* __restrict__ gate_w,  // [64, 1024]
    float* __restrict__ out_scores,    // [T, 2]
    float* __restrict__ out_idx,       // [T, 2] (written as float values)
    float* __restrict__ ws_partials,   // [numWG, 65]: 64 expert sums + z^2 sum
    int T)
{
    extern __shared__ float smem[];
    const int tid  = threadIdx.x;
    const int bid  = blockIdx.x;
    const int lane = tid & 31;
    const int wv   = tid >> 5;        // wave 0..7
    const int col  = lane & 15;
    const int half = lane >> 4;

    // ---- Stage gate_w -> LDS as [pair][e*2+parity] (one-time, L2-resident src)
    {
        const int e      = tid >> 2;              // 0..63
        const int jbase  = (tid & 3) * (KPAIRS/4);
        const float* gw  = gate_w + (size_t)e * H;
        #pragma unroll 4
        for (int i = 0; i < KPAIRS/4; ++i) {
            const int j = jbase + i;              // pair index
            v2f g = *(const v2f*)(gw + 2*j);
            *(v2f*)(smem + LDS_B + j*128 + 2*e) = g;
        }
    }
    __syncthreads();

    // ---- WMMA GEMM: this wave's 16 tokens x all 64 experts -----------------
    const int tok0 = (bid * WAVES + wv) * 16;
    const float* aRow = x + (size_t)(tok0 + col) * H + 2*half;

    v8f acc0 = {}, acc1 = {}, acc2 = {}, acc3 = {};
    #pragma unroll 8
    for (int k = 0; k < H; k += 4) {
        if ((k & 28) == 0)
            __builtin_prefetch(aRow + k + 1024, 0, 3);   // global_prefetch_b8
        v2f a = *(const v2f*)(aRow + k);
        const float* bb = smem + LDS_B + (k/2 + half)*128 + 2*col;
        v2f b0 = *(const v2f*)(bb +  0);
        v2f b1 = *(const v2f*)(bb + 32);
        v2f b2 = *(const v2f*)(bb + 64);
        v2f b3 = *(const v2f*)(bb + 96);
        acc0 = __builtin_amdgcn_wmma_f32_16x16x4_f32(false, a, false, b0, (short)0, acc0, false, false);
        acc1 = __builtin_amdgcn_wmma_f32_16x16x4_f32(false, a, false, b1, (short)0, acc1, false, false);
        acc2 = __builtin_amdgcn_wmma_f32_16x16x4_f32(false, a, false, b2, (short)0, acc2, false, false);
        acc3 = __builtin_amdgcn_wmma_f32_16x16x4_f32(false, a, false, b3, (short)0, acc3, false, false);
    }

    // ---- Spill logits to LDS: D[m][n] lives in reg r=m%8, lane=(m/8)*16+n --
    {
        float* srow = smem + LDS_S + wv * (16*EXPERTS);
        #pragma unroll
        for (int r = 0; r < 8; ++r) {
            const int m = r + 8*half;
            srow[m*EXPERTS +  0 + col] = acc0[r];
            srow[m*EXPERTS + 16 + col] = acc1[r];
            srow[m*EXPERTS + 32 + col] = acc2[r];
            srow[m*EXPERTS + 48 + col] = acc3[r];
        }
    }
    __syncthreads();

    // ---- Per-token softmax / top-2 / z  (token = tid for tid < 128) --------
    if (tid < TOK_WG) {
        float* srow = smem + LDS_S + tid * EXPERTS;   // wv*1024 + (tid%16)*64 == tid*64
        float mx = -INFINITY;
        #pragma unroll 8
        for (int e = 0; e < EXPERTS; ++e) mx = fmaxf(mx, srow[e]);

        float sum = 0.0f;
        float b1 = -INFINITY, b2 = -INFINITY;
        int   i1 = 0, i2 = 0;
        #pragma unroll 4
        for (int e = 0; e < EXPERTS; ++e) {
            const float l  = srow[e];
            const float ex = __expf(l - mx);
            srow[e] = ex;                 // keep unnormalized exp for load-balance
            sum += ex;
            if (l > b1)      { b2 = b1; i2 = i1; b1 = l; i1 = e; }
            else if (l > b2) { b2 = l;  i2 = e; }
        }
        const float inv = 1.0f / sum;
        const float z   = mx + __logf(sum);
        smem[LDS_Z   + tid] = z * z;
        smem[LDS_INV + tid] = inv;

        // reference: softmax over the top-2 *probabilities*
        const float p1 = __expf(b1 - mx) * inv;
        const float p2 = __expf(b2 - mx) * inv;
        const float eo = __expf(p2 - p1);
        const float s0 = 1.0f / (1.0f + eo);
        const float s1 = eo * s0;

        const int gt = bid * TOK_WG + tid;
        out_scores[2*gt + 0] = s0;
        out_scores[2*gt + 1] = s1;
        out_idx[2*gt + 0] = (float)i1;
        out_idx[2*gt + 1] = (float)i2;
    }
    __syncthreads();

    // ---- Deterministic per-WG partials: expert-load sums + z^2 sum ---------
    if (tid < EXPERTS) {
        float s = 0.0f;
        for (int t = 0; t < TOK_WG; ++t)
            s += smem[LDS_S + t*EXPERTS + tid] * smem[LDS_INV + t];
        ws_partials[bid * 65 + tid] = s;
    } else if (tid == EXPERTS) {
        float s = 0.0f;
        for (int t = 0; t < TOK_WG; ++t) s += smem[LDS_Z + t];
        ws_partials[bid * 65 + 64] = s;
    }
}

__global__ __launch_bounds__(64) void moe_loss_kernel(
    const float* __restrict__ ws_partials, float* __restrict__ out_loss,
    int numWG, int T)
{
    __shared__ float red[EXPERTS];
    const int e = threadIdx.x;
    float s = 0.0f;
    for (int w = 0; w < numWG; ++w) s += ws_partials[w * 65 + e];
    const float load = s / (float)T;
    const float d    = load - (1.0f / (float)EXPERTS);
    red[e] = d * d;
    __syncthreads();
    if (e == 0) {
        float lb = 0.0f;
        for (int i = 0; i < EXPERTS; ++i) lb += red[i];
        float zs = 0.0f;
        for (int w = 0; w < numWG; ++w) zs += ws_partials[w * 65 + 64];
        out_loss[0] = 0.01f * (float)EXPERTS * lb + 1e-4f * (zs / (float)T);
    }
}

extern "C" void kernel_launch(void* const* d_in, const int* in_sizes, int n_in,
                              void* d_out, int out_size, void* d_ws, size_t ws_size,
                              hipStream_t stream) {
    const float* x      = (const float*)d_in[0];
    const float* gate_w = (const float*)d_in[1];
    const int T = in_sizes[0] / H;          // 16384
    const int numWG = T / TOK_WG;           // 128

    float* outF       = (float*)d_out;
    float* out_scores = outF;               // [T,2]
    float* out_idx    = outF + 2 * T;       // [T,2] as float values
    float* out_loss   = outF + 4 * T;       // scalar
    float* partials   = (float*)d_ws;       // [numWG, 65]

    const size_t smem_bytes = (size_t)LDS_FLOATS * sizeof(float);  // ~289 KB, fits 320 KB WGP LDS
    moe_gate_kernel<<<numWG, 256, smem_bytes, stream>>>(
        x, gate_w, out_scores, out_idx, partials, T);
    moe_loss_kernel<<<1, 64, 0, stream>>>(partials, out_loss, numWG, T);
}